// HardContrastiveLossV6_V1_8993661517972
// MI455X (gfx1250) — compile-verified
//
#include <hip/hip_runtime.h>
#include <math.h>

typedef float v2f __attribute__((ext_vector_type(2)));
typedef float v8f __attribute__((ext_vector_type(8)));
typedef unsigned u4v __attribute__((ext_vector_type(4)));
typedef int i8v __attribute__((ext_vector_type(8)));
typedef int i4v __attribute__((ext_vector_type(4)));

#define TILE 128
#define KC   32
#define LDA  36   // padded LDS row stride (dwords): 32 data + 4 pad; even (8B-aligned b64), 36*r mod 64 distinct for r=0..15

// ---------------------------------------------------------------------------
// K1: hard_pos / hard_neg / norms / positive_pairs copy
// ---------------------------------------------------------------------------
__global__ __launch_bounds__(256) void k_prep(const float* __restrict__ emb,
                                              const int*   __restrict__ pairs,
                                              int*   __restrict__ pos_out,
                                              float* __restrict__ hard_pos,
                                              float* __restrict__ hard_neg,
                                              float* __restrict__ nrm,
                                              int B, int D) {
  const int b = blockIdx.x;
  const int t = threadIdx.x;
  // positive partner of row b (pairs are (2k,2k+1))
  const int pp = pairs[(b >> 1) * 2 + (1 - (b & 1))];
  // hard-neg partner from the reference's "last occurrence" scatter: B-1, except row B-1 -> B-2
  const int q  = (b == B - 1) ? (B - 2) : (B - 1);
  __shared__ float red[256];
  float ss = 0.f;
  for (int d = t; d < D; d += 256) {
    float e  = emb[(size_t)b  * D + d];
    float ep = emb[(size_t)pp * D + d];
    float eq = emb[(size_t)q  * D + d];
    hard_pos[(size_t)b * D + d] = 1.5f * e - 0.5f * ep;   // LAM_POS*e + (1-LAM_POS)*ep
    float h = 0.5f * (e + eq);                            // LAM_NEG*(e + e[partner])
    hard_neg[(size_t)b * D + d] = h;
    ss += h * h;
  }
  red[t] = ss;
  __syncthreads();
  for (int s = 128; s > 0; s >>= 1) { if (t < s) red[t] += red[t + s]; __syncthreads(); }
  if (t == 0) { nrm[b] = sqrtf(red[0]); pos_out[b] = pairs[b]; }
}

// ---------------------------------------------------------------------------
// K2: negative_pairs enumeration (row-major nonzero order). 134 MB, pure BW.
// ---------------------------------------------------------------------------
__global__ __launch_bounds__(256) void k_negpairs(int* __restrict__ out, int B) {
  const int i = blockIdx.x;
  const long long rowoff = (long long)i * (B - 1) - ((i + 1) >> 1);
  const int cnt = (B - 1) - (((i & 1) == 0) ? 1 : 0);
  int2* o = (int2*)out;
  for (int tt = threadIdx.x; tt < cnt; tt += 256) {
    int j;
    if ((i & 1) == 0) j = (tt < i) ? tt : tt + 2;  // skip {i, i+1}
    else              j = (tt < i) ? tt : tt + 1;  // skip {i}
    o[rowoff + tt] = make_int2(i, j);
  }
}

// ---------------------------------------------------------------------------
// TDM: one descriptor loads a TILE x KC f32 tile into LDS with hardware row
// padding reproducing the [TILE][LDA] layout (32 data dwords + 4 pad dwords).
// D# encoding per CDNA5 ISA ch.8 (group0 128b, group1 256b; groups 2/3 = 0).
// ---------------------------------------------------------------------------
__device__ __forceinline__ void tdm_load_tile(unsigned lds_byte, const float* gaddr,
                                              int tensor_d0, int tensor_d1) {
  const unsigned long long ga = (unsigned long long)(uintptr_t)gaddr;
  u4v g0;
  g0.x = 0x1u;                                            // count=1, user mode
  g0.y = lds_byte;                                        // lds_addr (bytes)
  g0.z = (unsigned)(ga & 0xFFFFFFFFu);                    // global_addr[31:0]
  g0.w = (unsigned)((ga >> 32) & 0x01FFFFFFu) | 0x80000000u; // addr[56:32] | type=2
  i8v g1;
  // data_size=2 (4B) | pad_enable | pad_interval=4 (32 dwords) | pad_amount=3 (4 dwords)
  g1[0] = (int)(0x20000u | 0x100000u | (4u << 22) | (3u << 25));
  g1[1] = (int)((unsigned)(tensor_d0 & 0xFFFF) << 16);    // tensor_dim0[15:0]
  g1[2] = (int)(((unsigned)tensor_d0 >> 16) | ((unsigned)(tensor_d1 & 0xFFFF) << 16));
  g1[3] = (int)(((unsigned)tensor_d1 >> 16) | ((unsigned)KC << 16));  // tile_dim0=KC
  g1[4] = TILE;                                           // tile_dim1=TILE, tile_dim2=0
  g1[5] = tensor_d0;                                      // tensor_dim0_stride[31:0]
  g1[6] = 0;                                              // stride hi / dim1_stride lo
  g1[7] = 0;
  const i4v gz4 = {0, 0, 0, 0};
  const i8v gz8 = {0, 0, 0, 0, 0, 0, 0, 0};
  __builtin_amdgcn_tensor_load_to_lds(g0, g1, gz4, gz4, gz8, 0);
}

// ---------------------------------------------------------------------------
// K3: sim = cos-sim(hard_neg) off-diag, 1.0 on diag.  S = H H^T via
// V_WMMA_F32_16X16X4_F32.  128x128 block tile, 8 waves of 32x64.
// Double-buffered LDS staging via the Tensor Data Mover (TENSORcnt).
// ---------------------------------------------------------------------------
__global__ __launch_bounds__(256) void k_simgemm(const float* __restrict__ H,
                                                 const float* __restrict__ nrm,
                                                 float* __restrict__ sim,
                                                 int B, int D) {
  __shared__ float As[2][TILE * LDA];
  __shared__ float Bs[2][TILE * LDA];
  const int row0 = blockIdx.y * TILE;
  const int col0 = blockIdx.x * TILE;
  const int t    = threadIdx.x;
  const int wave = t >> 5, lane = t & 31;
  const int wr = wave >> 1, wc = wave & 1;   // wave tile: rows wr*32, cols wc*64
  const int lm = lane & 15, lkh = lane >> 4; // lane row-in-16, K-half select

  const unsigned aBase = (unsigned)(unsigned long long)(uintptr_t)&As[0][0];
  const unsigned bBase = (unsigned)(unsigned long long)(uintptr_t)&Bs[0][0];
  const unsigned bufBytes = TILE * LDA * 4;

  v8f acc[2][4];
#pragma unroll
  for (int mi = 0; mi < 2; ++mi)
#pragma unroll
    for (int ni = 0; ni < 4; ++ni) acc[mi][ni] = (v8f)0.0f;

  const int nchunks = D / KC;
  if (wave == 0) {
    tdm_load_tile(aBase, H + (size_t)row0 * D, D, B);
    tdm_load_tile(bBase, H + (size_t)col0 * D, D, B);
  }

  for (int c = 0; c < nchunks; ++c) {
    const int cur = c & 1;
    if (wave == 0) {
      if (c + 1 < nchunks) {                 // prefetch next K-chunk
        const int k1 = (c + 1) * KC;
        const unsigned boff = (unsigned)(1 - cur) * bufBytes;
        tdm_load_tile(aBase + boff, H + (size_t)row0 * D + k1, D, B);
        tdm_load_tile(bBase + boff, H + (size_t)col0 * D + k1, D, B);
        __builtin_amdgcn_s_wait_tensorcnt(2); // chunk-c pair complete (in-order)
      } else {
        __builtin_amdgcn_s_wait_tensorcnt(0);
      }
    }
    __syncthreads();                          // publish chunk-c LDS data to all waves

#pragma unroll
    for (int kk = 0; kk < KC; kk += 4) {
      v2f a[2], b[4];
#pragma unroll
      for (int mi = 0; mi < 2; ++mi)
        a[mi] = *(const v2f*)(&As[cur][(wr * 32 + mi * 16 + lm) * LDA + kk + 2 * lkh]);
#pragma unroll
      for (int ni = 0; ni < 4; ++ni)
        b[ni] = *(const v2f*)(&Bs[cur][(wc * 64 + ni * 16 + lm) * LDA + kk + 2 * lkh]);
#pragma unroll
      for (int mi = 0; mi < 2; ++mi)
#pragma unroll
        for (int ni = 0; ni < 4; ++ni)
          acc[mi][ni] = __builtin_amdgcn_wmma_f32_16x16x4_f32(
              false, a[mi], false, b[ni], (short)0, acc[mi][ni], false, false);
    }
    __syncthreads();                          // buffer reusable for prefetch c+2
  }

  // Epilogue: D layout: vgpr r -> row = base + 8*lkh + r, col = base + lm
#pragma unroll
  for (int mi = 0; mi < 2; ++mi) {
    const int rbase = row0 + wr * 32 + mi * 16 + lkh * 8;
    float nr[8];
#pragma unroll
    for (int r = 0; r < 8; ++r) nr[r] = nrm[rbase + r];
#pragma unroll
    for (int ni = 0; ni < 4; ++ni) {
      const int gcol = col0 + wc * 64 + ni * 16 + lm;
      const float nc = nrm[gcol];
#pragma unroll
      for (int r = 0; r < 8; ++r) {
        const int grow = rbase + r;
        float s = acc[mi][ni][r] / fmaxf(nr[r] * nc, 1e-8f);
        if (grow == gcol) s = 1.0f;   // diagonal: original cosine sim of row with itself
        sim[(size_t)grow * B + gcol] = s;
      }
    }
  }
}

// ---------------------------------------------------------------------------
// K4: per-row loss: exp(sim/TAU), mask diag + positive partner, exact rank-3276
// threshold via 3-level LDS radix select on float bit patterns, sum >= thr.
// ---------------------------------------------------------------------------
__global__ __launch_bounds__(256) void k_rowloss(const float* __restrict__ sim,
                                                 const int*   __restrict__ pairs,
                                                 float* __restrict__ row_out,
                                                 int B) {
  const int r = blockIdx.x;
  const int t = threadIdx.x;
  const int pp = pairs[(r >> 1) * 2 + (1 - (r & 1))];
  __shared__ unsigned hist[2048];
  __shared__ unsigned coarse[256];
  __shared__ float    sred[256];
  __shared__ unsigned s_known;
  __shared__ int      s_rem;
  __shared__ float    s_pos;

  const float* row = sim + (size_t)r * B;
  float z[16];
#pragma unroll
  for (int i = 0; i < 16; ++i) {
    int c = t + 256 * i;
    float e = expf(row[c] * 5.0f);      // /TAU, TAU=0.2
    if (c == pp) s_pos = e;             // unmasked positive logit
    if (c == r || c == pp) e = 0.f;     // exp_m masking
    z[i] = e;
  }

  const int krank = (int)(0.8 * (double)(B - 1));  // 3276 for B=4096, frac==0
  unsigned known = 0;
  int kshift = 32;
  int remaining = krank;

  for (int lvl = 0; lvl < 3; ++lvl) {
    const int shift = (lvl == 0) ? 21 : (lvl == 1) ? 10 : 0;
    const int nb    = (lvl == 2) ? 1024 : 2048;
    for (int i = t; i < 2048; i += 256) hist[i] = 0;
    __syncthreads();
#pragma unroll
    for (int i = 0; i < 16; ++i) {
      unsigned bits = __float_as_uint(z[i]);
      bool cand = (kshift == 32) || ((bits >> kshift) == known);
      if (cand) atomicAdd(&hist[(bits >> shift) & (nb - 1)], 1u);
    }
    __syncthreads();
    const int chunk = nb / 256;
    unsigned cs = 0;
    for (int i = 0; i < chunk; ++i) cs += hist[t * chunk + i];
    coarse[t] = cs;
    __syncthreads();
    if (t == 0) {
      int rem = remaining;
      int ci = 0;
      while ((int)coarse[ci] <= rem) { rem -= coarse[ci]; ++ci; }
      int bin = ci * chunk;
      while ((int)hist[bin] <= rem) { rem -= hist[bin]; ++bin; }
      s_known = (kshift == 32) ? (unsigned)bin
                               : ((known << (kshift - shift)) | (unsigned)bin);
      s_rem = rem;
    }
    __syncthreads();
    known = s_known;
    remaining = s_rem;
    kshift = shift;
  }

  const float thr = __uint_as_float(known);   // exact rank-3276 value
  float loc = 0.f;
#pragma unroll
  for (int i = 0; i < 16; ++i)
    if (z[i] >= thr) loc += z[i];
  sred[t] = loc;
  __syncthreads();
  for (int s = 128; s > 0; s >>= 1) { if (t < s) sred[t] += sred[t + s]; __syncthreads(); }
  if (t == 0) {
    float p = s_pos;
    row_out[r] = logf((p + sred[0]) / p);     // -log(pos/(pos+sum))
  }
}

// ---------------------------------------------------------------------------
// K5: deterministic final reduction: loss = sum(row_loss)/B   (2P == B)
// ---------------------------------------------------------------------------
__global__ __launch_bounds__(256) void k_finloss(const float* __restrict__ row_loss,
                                                 float* __restrict__ out, int B) {
  __shared__ float red[256];
  int t = threadIdx.x;
  float s = 0.f;
  for (int i = t; i < B; i += 256) s += row_loss[i];
  red[t] = s;
  __syncthreads();
  for (int k = 128; k > 0; k >>= 1) { if (t < k) red[t] += red[t + k]; __syncthreads(); }
  if (t == 0) out[0] = red[0] / (float)B;
}

// ---------------------------------------------------------------------------
extern "C" void kernel_launch(void* const* d_in, const int* in_sizes, int n_in,
                              void* d_out, int out_size, void* d_ws, size_t ws_size,
                              hipStream_t stream) {
  const float* emb   = (const float*)d_in[0];
  const int*   pairs = (const int*)d_in[1];
  const int B = in_sizes[1];            // pairs is (B/2, 2) -> B ints; B = 4096
  const int D = in_sizes[0] / B;        // 256
  const int P = B / 2;
  const long long n_neg = (long long)B * (B - 1) - P;

  float* outf = (float*)d_out;
  long long off_pos  = 0;
  long long off_neg  = off_pos + B;
  long long off_hp   = off_neg + 2 * n_neg;
  long long off_hn   = off_hp + (long long)B * D;
  long long off_sim  = off_hn + (long long)B * D;
  long long off_loss = off_sim + (long long)B * B;

  int*   pos_out  = (int*)d_out;
  int*   neg_out  = (int*)(outf + off_neg);
  float* hard_pos = outf + off_hp;
  float* hard_neg = outf + off_hn;
  float* sim      = outf + off_sim;
  float* loss     = outf + off_loss;

  float* nrm      = (float*)d_ws;        // B floats
  float* row_loss = nrm + B;             // B floats

  k_prep<<<B, 256, 0, stream>>>(emb, pairs, pos_out, hard_pos, hard_neg, nrm, B, D);
  k_negpairs<<<B, 256, 0, stream>>>(neg_out, B);
  dim3 grid(B / TILE, B / TILE);
  k_simgemm<<<grid, 256, 0, stream>>>(hard_neg, nrm, sim, B, D);
  k_rowloss<<<B, 256, 0, stream>>>(sim, pairs, row_loss, B);
  k_finloss<<<1, 256, 0, stream>>>(row_loss, loss, B);
}